// _CenterLoss_68229850465030
// MI455X (gfx1250) — compile-verified
//
#include <hip/hip_runtime.h>

// ---------------------------------------------------------------------------
// CenterLoss on MI455X (gfx1250, wave32).
//
// Key identity: after the one-hot label mask + clamp, the loss reduces to
//   loss = (1/B) * sum_i clamp(||x_i||^2 + ||c_li||^2 - 2 x_i.c_li, 1e-12, 1e12)
//          + (C-1)*1e-12
// => no 4096x100000 GEMM; just 4096 gathered 64-dim distances (~2 MB traffic,
// latency-bound). The 16 dot products per 16-row tile are computed on the
// matrix pipe with V_WMMA_F32_16X16X4_F32 (exact f32), reading the diagonal
// of the 16x16 accumulator.
// ---------------------------------------------------------------------------

#define BATCH        4096
#define NUM_CLASSES  100000
#define FEAT         64
#define TILE_ROWS    16
#define NBLOCKS      (BATCH / TILE_ROWS)   // 256
#define CLAMP_MIN    1e-12f
#define CLAMP_MAX    1e12f

typedef __attribute__((ext_vector_type(2))) float v2f;
typedef __attribute__((ext_vector_type(8))) float v8f;

__global__ __launch_bounds__(32)
void center_loss_partial(const float* __restrict__ x,
                         const float* __restrict__ centers,
                         const int*   __restrict__ labels,
                         float*       __restrict__ partials) {
    const int lane = threadIdx.x;     // 0..31 (wave32)
    const int hi   = lane >> 4;       // 0: K%4 in {0,1}, 1: K%4 in {2,3}
    const int row  = lane & 15;       // row within the 16-row tile
    const int r0   = blockIdx.x * TILE_ROWS;

    const int lab = labels[r0 + row];
    const float* __restrict__ xr = x       + (size_t)(r0 + row) * FEAT;
    const float* __restrict__ cr = centers + (size_t)lab        * FEAT;

    // Two interleaved accumulators to pipeline the XDL dependency chain.
    v8f acc0 = {};
    v8f acc1 = {};
    float xsq = 0.0f, csq = 0.0f;

    // A (16x4 f32): lane holds A[row][4k+2*hi .. +1] in 2 VGPRs.
    // B (4x16 f32): lane holds B[4k+2*hi .. +1][row]  in 2 VGPRs.
    // Both map to a contiguous float2 at column 4k + 2*hi of the row.
#pragma unroll
    for (int k = 0; k < FEAT / 4; k += 2) {
        const int col0 = 4 * k + 2 * hi;
        const int col1 = 4 * (k + 1) + 2 * hi;
        v2f a0 = *(const v2f*)(xr + col0);
        v2f b0 = *(const v2f*)(cr + col0);
        v2f a1 = *(const v2f*)(xr + col1);
        v2f b1 = *(const v2f*)(cr + col1);
        xsq += a0.x * a0.x + a0.y * a0.y + a1.x * a1.x + a1.y * a1.y;
        csq += b0.x * b0.x + b0.y * b0.y + b1.x * b1.x + b1.y * b1.y;
        acc0 = __builtin_amdgcn_wmma_f32_16x16x4_f32(
            false, a0, false, b0, (short)0, acc0, false, false);
        acc1 = __builtin_amdgcn_wmma_f32_16x16x4_f32(
            false, a1, false, b1, (short)0, acc1, false, false);
    }
    v8f acc = acc0 + acc1;

    // Lane L and lane L^16 covered complementary halves of row L&15.
    xsq += __shfl_xor(xsq, 16, 32);
    csq += __shfl_xor(csq, 16, 32);

    // Diagonal (n,n) of the 16x16 f32 C/D tile:
    //   n in [0,8):  lane n,      VGPR n
    //   n in [8,16): lane n + 16, VGPR n - 8
    // => active lanes {0..7, 24..31}, VGPR index = lane & 7.
    const int idx = lane & 7;
    float s0 = (idx & 1) ? acc[1] : acc[0];
    float s1 = (idx & 1) ? acc[3] : acc[2];
    float s2 = (idx & 1) ? acc[5] : acc[4];
    float s3 = (idx & 1) ? acc[7] : acc[6];
    float t0 = (idx & 2) ? s1 : s0;
    float t1 = (idx & 2) ? s3 : s2;
    float dot = (idx & 4) ? t1 : t0;

    const bool isDiag = (lane < 8) || (lane >= 24);
    float dist = xsq + csq - 2.0f * dot;             // reference addmm identity
    dist = fminf(fmaxf(dist, CLAMP_MIN), CLAMP_MAX); // clamp per element
    float v = isDiag ? dist : 0.0f;

    // Deterministic full-wave reduction (fixed shuffle order).
#pragma unroll
    for (int off = 16; off >= 1; off >>= 1)
        v += __shfl_xor(v, off, 32);

    if (lane == 0)
        partials[blockIdx.x] = v;
}

__global__ __launch_bounds__(32)
void center_loss_final(const float* __restrict__ partials,
                       float*       __restrict__ out) {
    const int lane = threadIdx.x;   // single wave
    float s = 0.0f;
#pragma unroll
    for (int i = 0; i < NBLOCKS / 32; ++i)
        s += partials[lane + 32 * i];
#pragma unroll
    for (int off = 16; off >= 1; off >>= 1)
        s += __shfl_xor(s, off, 32);
    if (lane == 0) {
        // (C-1) masked zeros per row each clamp to 1e-12; B cancels.
        out[0] = s / (float)BATCH + (float)(NUM_CLASSES - 1) * CLAMP_MIN;
    }
}

extern "C" void kernel_launch(void* const* d_in, const int* in_sizes, int n_in,
                              void* d_out, int out_size, void* d_ws, size_t ws_size,
                              hipStream_t stream) {
    const float* x       = (const float*)d_in[0];   // (4096, 64) f32
    const float* centers = (const float*)d_in[1];   // (100000, 64) f32
    const int*   labels  = (const int*)  d_in[2];   // (4096,) int
    float*       out     = (float*)d_out;           // scalar f32
    float*       parts   = (float*)d_ws;            // 256 f32 partials

    center_loss_partial<<<NBLOCKS, 32, 0, stream>>>(x, centers, labels, parts);
    center_loss_final<<<1, 32, 0, stream>>>(parts, out);
}